// DualTransformerLayer_39986145525886
// MI455X (gfx1250) — compile-verified
//
#include <hip/hip_runtime.h>
#include <math.h>

// ---------------------------------------------------------------------------
// MI455X (gfx1250) implementation. wave32. All dense math on V_WMMA_F32_16X16X4_F32
// (fp32 matrix path -> preserves fp32 reference numerics; this workload is
// matrix-core bound: ~260 GFLOP vs ~13us of HBM traffic at 23.3 TB/s).
// GEMMs stage A/B tiles through LDS in WMMA fragment layout (v2f pairs) so the
// inner loop is ds_load_b64 x3 + v_wmma x2 per K-quad with zero branches.
// ---------------------------------------------------------------------------

typedef float v2f __attribute__((ext_vector_type(2)));
typedef float v8f __attribute__((ext_vector_type(8)));

#define GF_GELU   1
#define GF_ACC    2
#define GF_BCASTP 4

// constants: B=32 N=379 H=512 NH=8 P=3 WGH=4, BN=12128, BPN=36384, PN=1137

__device__ __forceinline__ v8f wmma_f32_16x16x4(v2f a, v2f b, v8f c) {
  // 8 args: (neg_a, A, neg_b, B, c_mod, C, reuse_a, reuse_b)
  return __builtin_amdgcn_wmma_f32_16x16x4_f32(false, a, false, b, (short)0, c,
                                               false, false);
}

__device__ __forceinline__ float gelu_exact(float x) {
  return 0.5f * x * (1.0f + erff(x * 0.7071067811865475f));
}

// ---------------------------------------------------------------------------
// Generic WMMA-f32 GEMM:  C[M,N] = act( rs*A @ W + bias + (ACC ? C : 0) )
// A row-major [M,K] lda, W row-major [K,N] ldw (ldw % 4 == 0 required).
// Block = 128 thr (4 waves), tile 32(M) x 64(N) x 16(K) staged via LDS.
// LDS layout packs WMMA fragments directly:
//   As[kp][m] = {A[m][kb+2kp], A[m][kb+2kp+1]}   (kp = 0..7)
//   Bs[kp][n] = {W[kb+2kp][n], W[kb+2kp+1][n]}
// Fragment reads: a = As[2kk+half][m], b = Bs[2kk+half][n]  (one ds_load_b64).
// Row strides 48/80 v2f => row stride == 32 banks mod 64 -> conflict-free.
// ---------------------------------------------------------------------------
__global__ __launch_bounds__(128) void k_gemm(
    const float* __restrict__ A, int lda,
    const float* __restrict__ W, int ldw,
    const float* __restrict__ bias,
    const float* __restrict__ rowscale, int rsdiv,
    float* __restrict__ C, int ldc,
    int M, int K, int N, int flags)
{
  __shared__ v2f As[8][48];
  __shared__ v2f Bs[8][80];

  const int t = threadIdx.x;
  const int lane = t & 31, wave = t >> 5;
  const int wm = wave & 1, wn = wave >> 1;
  const int lm = lane & 15, half = lane >> 4;
  const int mBase = blockIdx.y * 32;
  const int nBase = blockIdx.x * 64;
  if (mBase >= M) return;   // uniform per block

  // ---- staging coordinates ----
  const int sm  = t >> 2;          // 0..31 : A row within tile
  const int sk4 = (t & 3) * 4;     // 0,4,8,12 : A k-quad
  const int sbk = t >> 3;          // 0..15 : B k-row within tile
  const int sbn = (t & 7) * 8;     // 0..56 : B col group (8 floats)

  // A row pointer with broadcast map / row-scale / m-guard folded in
  const int gm  = mBase + sm;
  const int gmc = gm < M ? gm : M - 1;
  long arow = gmc;
  if (flags & GF_BCASTP) arow = (long)(gmc / 1137) * 379 + (gmc % 379);
  float ascale = (gm < M) ? 1.f : 0.f;
  if (rowscale) ascale *= rowscale[gmc / rsdiv];
  const float* __restrict__ Ar = A + arow * (long)lda;

  // ---- fragment coordinates ----
  const int fm = wm * 16 + lm;           // A fragment row (local)
  const int n0 = wn * 32 + lm, n1 = n0 + 16;
  const int gn0 = nBase + n0, gn1 = nBase + n1;

  v8f c0 = {}; v8f c1 = {};
  for (int kb = 0; kb < K; kb += 16) {
    // ---- stage A (32x16): unconditional clamped loads + select zero ----
    {
      const int kg = kb + sk4;
      const int kc0 = kg     < K ? kg     : K - 1;
      const int kc1 = kg + 1 < K ? kg + 1 : K - 1;
      const int kc2 = kg + 2 < K ? kg + 2 : K - 1;
      const int kc3 = kg + 3 < K ? kg + 3 : K - 1;
      float v0 = Ar[kc0], v1 = Ar[kc1], v2 = Ar[kc2], v3 = Ar[kc3];
      v0 = (kg     < K ? v0 : 0.f) * ascale;
      v1 = (kg + 1 < K ? v1 : 0.f) * ascale;
      v2 = (kg + 2 < K ? v2 : 0.f) * ascale;
      v3 = (kg + 3 < K ? v3 : 0.f) * ascale;
      const int kp = sk4 >> 1;
      v2f p0; p0.x = v0; p0.y = v1;
      v2f p1; p1.x = v2; p1.y = v3;
      As[kp][sm]     = p0;
      As[kp + 1][sm] = p1;
    }
    // ---- stage B (16x64): aligned float4 loads from one (clamped) row ----
    {
      const int kg = kb + sbk;
      const int kc = kg < K ? kg : K - 1;
      const float kf = kg < K ? 1.f : 0.f;
      const float4* wr = (const float4*)(W + (size_t)kc * ldw + nBase + sbn);
      const float4 q0 = wr[0], q1 = wr[1];
      const int kp = sbk >> 1;
      const int comp = sbk & 1;
      float* brow = (float*)&Bs[kp][0];
      brow[(sbn + 0) * 2 + comp] = q0.x * kf;
      brow[(sbn + 1) * 2 + comp] = q0.y * kf;
      brow[(sbn + 2) * 2 + comp] = q0.z * kf;
      brow[(sbn + 3) * 2 + comp] = q0.w * kf;
      brow[(sbn + 4) * 2 + comp] = q1.x * kf;
      brow[(sbn + 5) * 2 + comp] = q1.y * kf;
      brow[(sbn + 6) * 2 + comp] = q1.z * kf;
      brow[(sbn + 7) * 2 + comp] = q1.w * kf;
    }
    // prefetch next K-tile (speculative; lowers to global_prefetch_b8)
    __builtin_prefetch(Ar + kb + 16 + sk4, 0, 0);
    __builtin_prefetch(W + (size_t)(kb + 16 + sbk) * ldw + nBase + sbn, 0, 0);

    __syncthreads();
#pragma unroll
    for (int kk = 0; kk < 4; ++kk) {
      const v2f a  = As[kk * 2 + half][fm];
      const v2f b0 = Bs[kk * 2 + half][n0];
      const v2f b1 = Bs[kk * 2 + half][n1];
      c0 = wmma_f32_16x16x4(a, b0, c0);
      c1 = wmma_f32_16x16x4(a, b1, c1);
    }
    __syncthreads();
  }

  const float bv0 = bias ? bias[gn0] : 0.f;
  const float bv1 = bias ? bias[gn1] : 0.f;
#pragma unroll
  for (int r = 0; r < 8; ++r) {
    const int row = mBase + wm * 16 + r + 8 * half;
    if (row >= M) continue;
    const long o0 = (long)row * ldc + gn0;
    const long o1 = (long)row * ldc + gn1;
    float x0 = c0[r] + bv0;
    float x1 = c1[r] + bv1;
    if (flags & GF_ACC) { x0 += C[o0]; x1 += C[o1]; }
    if (flags & GF_GELU) { x0 = gelu_exact(x0); x1 = gelu_exact(x1); }
    C[o0] = x0; C[o1] = x1;
  }
}

// ---------------------------------------------------------------------------
// Fused attention: block = (b,head) x 16-row tile of queries; 128 thr (4 waves)
// s = (q@k^T)*scale*(1+mask); softmax over 379 keys in LDS; ctx = p@v; writes
// attn_out = x + ctx + fused_nh.  Both matmuls via WMMA f32.
// ---------------------------------------------------------------------------
__global__ __launch_bounds__(128) void k_attn(
    const float* __restrict__ qkv,   // (B,N,3H)
    const float* __restrict__ mask,  // (B,N,N)
    const float* __restrict__ x,     // (B,N,H)
    const float* __restrict__ fn,    // (B,N,H) fused_nh
    float* __restrict__ ao)          // (B,N,H)
{
  __shared__ float sc[16][388];      // stride 388: 388%64=4 -> conflict-free cols
  __shared__ float red[16][8];
  __shared__ float rowmax[16];
  __shared__ float rowsum[16];

  const int bh = blockIdx.x;
  const int b = bh >> 3, h = bh & 7;
  const int iBase = blockIdx.y * 16;
  const int lane = threadIdx.x & 31, wave = threadIdx.x >> 5;
  const int lm = lane & 15, half = lane >> 4;
  const float* qkvb = qkv + (size_t)b * 379 * 1536 + h * 64;

  // ---- scores: q(16x64) @ k^T(64x16-tiles), 24 j-tiles over 4 waves ----
  {
    const int irow = iBase + lm;
    const int irc = irow < 379 ? irow : 378;
    const float qf = irow < 379 ? 1.f : 0.f;
    const float* qp = qkvb + (size_t)irc * 1536 + 2 * half;
    for (int jt = wave; jt < 24; jt += 4) {
      const int jcol = jt * 16 + lm;
      const int jcc = jcol < 379 ? jcol : 378;
      const float kf = jcol < 379 ? 1.f : 0.f;
      const float* kp = qkvb + 512 + (size_t)jcc * 1536 + 2 * half;
      v8f c = {};
#pragma unroll
      for (int kb = 0; kb < 64; kb += 4) {
        v2f a, bf;
        a.x  = qp[kb] * qf;  a.y  = qp[kb + 1] * qf;
        bf.x = kp[kb] * kf;  bf.y = kp[kb + 1] * kf;
        c = wmma_f32_16x16x4(a, bf, c);
      }
#pragma unroll
      for (int r = 0; r < 8; ++r) {
        const int il = r + 8 * half;
        const int ig = iBase + il;
        const int im = ig < 379 ? ig : 378;
        // branch-free: always load (clamped), then select
        const float mk = mask[((size_t)b * 379 + im) * 379 + jcc];
        const float sv = c[r] * 0.125f * (1.f + mk);
        sc[il][jcol] = (jcol < 379) ? sv : -1e30f;   // pad -> prob 0
      }
    }
  }
  __syncthreads();

  // ---- row softmax (unnormalized exp + rowsum), 8 threads per row ----
  {
    const int t = threadIdx.x;
    const int rr = t >> 3, sl = t & 7;
    float mx = -3e38f;
    for (int j = sl; j < 379; j += 8) mx = fmaxf(mx, sc[rr][j]);
    red[rr][sl] = mx; __syncthreads();
    if (sl == 0) {
      float m2 = red[rr][0];
      for (int u = 1; u < 8; ++u) m2 = fmaxf(m2, red[rr][u]);
      rowmax[rr] = m2;
    }
    __syncthreads();
    const float rm = rowmax[rr];
    float s = 0.f;
    for (int j = sl; j < 384; j += 8) {
      const float e = expf(sc[rr][j] - rm);
      sc[rr][j] = e; s += e;
    }
    red[rr][sl] = s; __syncthreads();
    if (sl == 0) {
      float s2 = 0.f;
      for (int u = 0; u < 8; ++u) s2 += red[rr][u];
      rowsum[rr] = s2;
    }
    __syncthreads();
  }

  // ---- ctx: probs(16x384) @ v(384x64); each wave owns a 16-col slab ----
  {
    const int col = wave * 16 + lm;
    const float* vp = qkvb + 1024 + col;
    v8f c = {};
    for (int kb = 0; kb < 384; kb += 4) {
      const int ka = kb + 2 * half;
      v2f a, bf;
      a.x = sc[lm][ka]; a.y = sc[lm][ka + 1];            // pads are exp()=0
      const int k0c = ka < 379 ? ka : 0;           const float f0 = ka < 379 ? 1.f : 0.f;
      const int k1c = (ka + 1) < 379 ? ka + 1 : 0; const float f1 = (ka + 1) < 379 ? 1.f : 0.f;
      bf.x = vp[(size_t)k0c * 1536] * f0;
      bf.y = vp[(size_t)k1c * 1536] * f1;
      c = wmma_f32_16x16x4(a, bf, c);
    }
#pragma unroll
    for (int r = 0; r < 8; ++r) {
      const int il = r + 8 * half;
      const int ig = iBase + il;
      if (ig < 379) {
        const size_t o = ((size_t)b * 379 + ig) * 512 + h * 64 + col;
        ao[o] = x[o] + c[r] / rowsum[il] + fn[o];
      }
    }
  }
}

// ---------------------------------------------------------------------------
// LayerNorm: one row (H=512) per block of 256 threads.
// ---------------------------------------------------------------------------
__global__ void k_layernorm(const float* __restrict__ x,
                            const float* __restrict__ g,
                            const float* __restrict__ bb,
                            float* __restrict__ out)
{
  __shared__ float s1[256], s2[256];
  const int row = blockIdx.x, t = threadIdx.x;
  const float* xr = x + (size_t)row * 512;
  const float v0 = xr[t], v1 = xr[t + 256];
  s1[t] = v0 + v1; s2[t] = v0 * v0 + v1 * v1;
  __syncthreads();
  for (int s = 128; s > 0; s >>= 1) {
    if (t < s) { s1[t] += s1[t + s]; s2[t] += s2[t + s]; }
    __syncthreads();
  }
  const float mean = s1[0] * (1.f / 512.f);
  const float var  = s2[0] * (1.f / 512.f) - mean * mean;
  const float rstd = rsqrtf(var + 1e-5f);
  float* o = out + (size_t)row * 512;
  o[t]       = (v0 - mean) * rstd * g[t]       + bb[t];
  o[t + 256] = (v1 - mean) * rstd * g[t + 256] + bb[t + 256];
}

// ---------------------------------------------------------------------------
// Per-(b,p): cosine sim(conn, prior) and 1/sum|prior|.
// ---------------------------------------------------------------------------
__global__ void k_prior_stats(const float* __restrict__ conn,
                              const float* __restrict__ pri,
                              float* __restrict__ sims,
                              float* __restrict__ invmass)
{
  __shared__ float r0[256], r1[256], r2[256], r3[256];
  const int bp = blockIdx.x, b = bp / 3, t = threadIdx.x;
  const float* c = conn + (size_t)b * 143641;
  const float* q = pri + (size_t)bp * 143641;
  float dot = 0, nq = 0, nc = 0, m = 0;
  for (int i = t; i < 143641; i += 256) {
    const float cv = c[i], qv = q[i];
    dot += cv * qv; nq += qv * qv; nc += cv * cv; m += fabsf(qv);
  }
  r0[t] = dot; r1[t] = nq; r2[t] = nc; r3[t] = m;
  __syncthreads();
  for (int s = 128; s > 0; s >>= 1) {
    if (t < s) { r0[t] += r0[t+s]; r1[t] += r1[t+s]; r2[t] += r2[t+s]; r3[t] += r3[t+s]; }
    __syncthreads();
  }
  if (t == 0) {
    sims[bp] = r0[0] / (fmaxf(sqrtf(r2[0]), 1e-12f) * fmaxf(sqrtf(r1[0]), 1e-12f));
    invmass[bp] = 1.f / r3[0];
  }
}

__global__ void k_sw(const float* __restrict__ sims, float* __restrict__ sw) {
  const int b = threadIdx.x;
  if (b >= 32) return;
  const float s0 = sims[b*3], s1 = sims[b*3+1], s2 = sims[b*3+2];
  const float mx = fmaxf(s0, fmaxf(s1, s2));
  const float e0 = expf(s0-mx), e1 = expf(s1-mx), e2 = expf(s2-mx);
  const float inv = 1.f / (e0 + e1 + e2);
  sw[b*3] = e0*inv; sw[b*3+1] = e1*inv; sw[b*3+2] = e2*inv;
}

__global__ void k_fused_nh(const float* __restrict__ sw,
                           const float* __restrict__ pe,
                           float* __restrict__ fn)
{
  const size_t idx = (size_t)blockIdx.x * 256 + threadIdx.x;
  if (idx >= (size_t)6209536) return;
  const int h = (int)(idx % 512);
  const size_t bn = idx / 512;
  const int n = (int)(bn % 379);
  const int b = (int)(bn / 379);
  const float* s = sw + b * 3;
  const size_t base = ((size_t)(b * 3) * 379 + n) * 512 + h;
  fn[idx] = s[0]*pe[base] + s[1]*pe[base + (size_t)379*512] + s[2]*pe[base + (size_t)2*379*512];
}

__global__ void k_tokens(const float* __restrict__ f2, float* __restrict__ tok) {
  const int bp = blockIdx.x, t = threadIdx.x;
  for (int h = t; h < 512; h += 256) {
    float s = 0.f;
    const float* p = f2 + ((size_t)bp * 379) * 512 + h;
    for (int n = 0; n < 379; ++n) s += p[(size_t)n * 512];
    tok[bp * 512 + h] = s * (1.f / 379.f);
  }
}

// tiny weight-gen MHA over P=3 tokens, WGH=4 heads, hd=128; one thread per b
__global__ void k_wg_attn(const float* __restrict__ qw,
                          float* __restrict__ aw_out,
                          float* __restrict__ msum)
{
  const int b = threadIdx.x;
  if (b >= 32) return;
  const float* base = qw + (size_t)b * 3 * 1536;
  float aw[3] = {0.f, 0.f, 0.f};
  for (int hh = 0; hh < 4; ++hh) {
    for (int i = 0; i < 3; ++i) {
      float scv[3];
      for (int j = 0; j < 3; ++j) {
        float d = 0.f;
        const float* qp = base + i * 1536 + hh * 128;
        const float* kp = base + j * 1536 + 512 + hh * 128;
        for (int u = 0; u < 128; ++u) d += qp[u] * kp[u];
        scv[j] = d * 0.08838834764831845f;   // 1/sqrt(128)
      }
      const float mx = fmaxf(scv[0], fmaxf(scv[1], scv[2]));
      const float e0 = expf(scv[0]-mx), e1 = expf(scv[1]-mx), e2 = expf(scv[2]-mx);
      const float inv = 1.f / (e0 + e1 + e2);
      aw[0] += e0*inv; aw[1] += e1*inv; aw[2] += e2*inv;
    }
  }
  const float a0 = aw[0]*(1.f/12.f), a1 = aw[1]*(1.f/12.f), a2 = aw[2]*(1.f/12.f);
  const float mx = fmaxf(a0, fmaxf(a1, a2));
  const float e0 = expf((a0-mx)*100.f), e1 = expf((a1-mx)*100.f), e2 = expf((a2-mx)*100.f);
  const float s = e0 + e1 + e2;
  aw_out[b*3]   = e0/s; aw_out[b*3+1] = e1/s; aw_out[b*3+2] = e2/s;
  msum[b] = (e0 + e1 + e2) / s;
}

__global__ void k_mask(const float* __restrict__ conn,
                       const float* __restrict__ msum,
                       const float* __restrict__ alpha,
                       float* __restrict__ mask)
{
  const size_t idx = (size_t)blockIdx.x * 256 + threadIdx.x;
  if (idx >= (size_t)4596512) return;
  const int b = (int)(idx / 143641);
  const float a = alpha[0];
  mask[idx] = tanhf(0.5f * a * msum[b] + (1.f - 0.5f * a) * conn[idx]);
}

__global__ void k_xnew(const float* __restrict__ ao,
                       const float* __restrict__ ffn,
                       float* __restrict__ out)
{
  const size_t idx = (size_t)blockIdx.x * 256 + threadIdx.x;
  if (idx >= (size_t)6209536) return;
  out[idx] = ao[idx] + ffn[idx];
}

__device__ float blk_reduce(float v, float* sm, bool ismax) {
  const int t = threadIdx.x;
  sm[t] = v; __syncthreads();
  for (int s = 128; s > 0; s >>= 1) {
    if (t < s) sm[t] = ismax ? fmaxf(sm[t], sm[t + s]) : (sm[t] + sm[t + s]);
    __syncthreads();
  }
  const float r = sm[0]; __syncthreads();
  return r;
}

__global__ void k_distill(const float* __restrict__ fF, const float* __restrict__ xF,
                          const float* __restrict__ fS, const float* __restrict__ xS,
                          const float* __restrict__ dw, float* __restrict__ acc)
{
  __shared__ float sm[256];
  const int row = blockIdx.x, t = threadIdx.x;
  const size_t base = (size_t)row * 512;
  const float wf = dw[0], wsv = dw[1];
  const float pF0 = wf * fF[base+t]     + (1.f - wf) * xF[base+t];
  const float pF1 = wf * fF[base+t+256] + (1.f - wf) * xF[base+t+256];
  const float pS0 = wsv * fS[base+t]     + (1.f - wsv) * xS[base+t];
  const float pS1 = wsv * fS[base+t+256] + (1.f - wsv) * xS[base+t+256];
  const float mF = blk_reduce(fmaxf(pF0, pF1), sm, true);
  const float sF = blk_reduce(expf(pF0-mF) + expf(pF1-mF), sm, false);
  const float lseF = mF + logf(sF);
  const float mS = blk_reduce(fmaxf(pS0, pS1), sm, true);
  const float sS = blk_reduce(expf(pS0-mS) + expf(pS1-mS), sm, false);
  const float lseS = mS + logf(sS);
  const float lF0 = pF0-lseF, lF1 = pF1-lseF, lS0 = pS0-lseS, lS1 = pS1-lseS;
  float t1 = expf(lS0)*(lS0-lF0) + expf(lS1)*(lS1-lF1);
  float t2 = expf(lF0)*(lF0-lS0) + expf(lF1)*(lF1-lS1);
  t1 = blk_reduce(t1, sm, false);
  t2 = blk_reduce(t2, sm, false);
  if (t == 0) { atomicAdd(&acc[0], t1); atomicAdd(&acc[1], t2); }
}

__global__ void k_zero2(float* acc) { if (threadIdx.x < 2) acc[threadIdx.x] = 0.f; }

__global__ void k_finalize(const float* __restrict__ acc, float* __restrict__ out) {
  if (threadIdx.x == 0) out[0] = 0.5f * (acc[0] + acc[1]) * (1.f / 12128.f);
}

// ---------------------------------------------------------------------------
extern "C" void kernel_launch(void* const* d_in, const int* in_sizes, int n_in,
                              void* d_out, int out_size, void* d_ws, size_t ws_size,
                              hipStream_t stream)
{
  const float* x_in[2] = {(const float*)d_in[0], (const float*)d_in[1]};
  const float* pri[2]  = {(const float*)d_in[2], (const float*)d_in[3]};
  const float* conn[2] = {(const float*)d_in[4], (const float*)d_in[5]};
  const float* dw      = (const float*)d_in[6];
  const float* wqkv[2] = {(const float*)d_in[7], (const float*)d_in[8]};
  const float* ln_g    = (const float*)d_in[9];
  const float* ln_b    = (const float*)d_in[10];
  const float* ffn_w1  = (const float*)d_in[11];
  const float* ffn_b1  = (const float*)d_in[12];
  const float* ffn_w2  = (const float*)d_in[13];
  const float* ffn_b2  = (const float*)d_in[14];
  const float* pw      = (const float*)d_in[15];
  const float* pb      = (const float*)d_in[16];
  const float* f1w     = (const float*)d_in[17];
  const float* f1b     = (const float*)d_in[18];
  const float* f2w     = (const float*)d_in[19];
  const float* f2b     = (const float*)d_in[20];
  const float* ipw     = (const float*)d_in[21];
  const float* ipb     = (const float*)d_in[22];
  const float* alpha   = (const float*)d_in[23];

  float* ws = (float*)d_ws;
  size_t off = 0;
  auto alloc = [&](size_t n) { float* p = ws + off; off += n; return p; };
  const size_t ZN = (size_t)12128 * 512;             // B*N*H
  float* zbuf  = alloc(ZN);
  float* qkv   = alloc((size_t)12128 * 1536);
  float* pe    = alloc((size_t)36384 * 512);         // later reused as fused2
  float* h1    = alloc((size_t)36384 * 512);
  float* fn    = alloc(ZN);
  float* ao    = alloc(ZN);
  float* ffh   = alloc((size_t)12128 * 2048);
  float* ffnb0 = alloc(ZN);
  float* ffnb1 = alloc(ZN);
  float* maskb = alloc((size_t)32 * 379 * 379);
  float* tok   = alloc((size_t)96 * 512);
  float* qw    = alloc((size_t)96 * 1536);
  float* sims  = alloc(96);
  float* invm  = alloc(96);
  float* swv   = alloc(96);
  float* msum  = alloc(32);
  float* acc   = alloc(2);
  float* ffnb[2] = {ffnb0, ffnb1};

  float* out = (float*)d_out;
  const size_t X = ZN;
  float* out_x[2]  = {out, out + X};
  float* out_loss  = out + 2 * X;
  float* out_aw[2] = {out + 2 * X + 1, out + 2 * X + 1 + 96};

  for (int s = 0; s < 2; ++s) {
    // z = LN(x)
    k_layernorm<<<12128, 256, 0, stream>>>(x_in[s], ln_g + s * 512, ln_b + s * 512, zbuf);
    // qkv = z @ w_qkv
    k_gemm<<<dim3(24, 379), 128, 0, stream>>>(zbuf, 512, wqkv[s], 1536,
        nullptr, nullptr, 1, qkv, 1536, 12128, 512, 1536, 0);
    // cosine sims + 1/mass per (b,p); sw = softmax(sims)
    k_prior_stats<<<96, 256, 0, stream>>>(conn[s], pri[s], sims, invm);
    k_sw<<<1, 32, 0, stream>>>(sims, swv);
    // pe = (priors/mass) @ pw + pb   (row-scale folded into GEMM; K=379 padded)
    k_gemm<<<dim3(8, 1137), 128, 0, stream>>>(pri[s], 379, pw, 512,
        pb, invm, 379, pe, 512, 36384, 379, 512, 0);
    // fused_nh = sum_p sw * pe
    k_fused_nh<<<(int)((ZN + 255) / 256), 256, 0, stream>>>(swv, pe, fn);
    // edge_fuser: h1 = pe @ f1w[:H]; h1 = gelu(z_bcast @ f1w[H:] + f1b + h1)
    k_gemm<<<dim3(8, 1137), 128, 0, stream>>>(pe, 512, f1w, 512,
        nullptr, nullptr, 1, h1, 512, 36384, 512, 512, 0);
    k_gemm<<<dim3(8, 1137), 128, 0, stream>>>(zbuf, 512, f1w + (size_t)512 * 512, 512,
        f1b, nullptr, 1, h1, 512, 36384, 512, 512, GF_ACC | GF_GELU | GF_BCASTP);
    // fused2 = h1 @ f2w + f2b  (reuses pe buffer)
    k_gemm<<<dim3(8, 1137), 128, 0, stream>>>(h1, 512, f2w, 512,
        f2b, nullptr, 1, pe, 512, 36384, 512, 512, 0);
    // tokens = mean_n(fused2); in-proj; tiny MHA -> aw_, msum
    k_tokens<<<96, 256, 0, stream>>>(pe, tok);
    k_gemm<<<dim3(24, 3), 128, 0, stream>>>(tok, 512, ipw, 1536,
        ipb, nullptr, 1, qw, 1536, 96, 512, 1536, 0);
    k_wg_attn<<<1, 32, 0, stream>>>(qw, out_aw[s], msum);
    // mask = tanh(a/2*msum + (1-a/2)*conn)
    k_mask<<<(4596512 + 255) / 256, 256, 0, stream>>>(conn[s], msum, alpha, maskb);
    // attention (fused residual: ao = x + ctx + fused_nh)
    k_attn<<<dim3(256, 24), 128, 0, stream>>>(qkv, maskb, x_in[s], fn, ao);
    // FFN
    k_layernorm<<<12128, 256, 0, stream>>>(ao, ln_g + (2 + s) * 512, ln_b + (2 + s) * 512, zbuf);
    k_gemm<<<dim3(32, 379), 128, 0, stream>>>(zbuf, 512, ffn_w1 + (size_t)s * 512 * 2048, 2048,
        ffn_b1 + s * 2048, nullptr, 1, ffh, 2048, 12128, 512, 2048, GF_GELU);
    k_gemm<<<dim3(8, 379), 128, 0, stream>>>(ffh, 2048, ffn_w2 + (size_t)s * 2048 * 512, 512,
        ffn_b2 + s * 512, nullptr, 1, ffnb[s], 512, 12128, 2048, 512, 0);
    // x_new = attn_out + ffn
    k_xnew<<<(int)((ZN + 255) / 256), 256, 0, stream>>>(ao, ffnb[s], out_x[s]);
  }
  // symmetric KL distillation loss
  k_zero2<<<1, 32, 0, stream>>>(acc);
  k_distill<<<12128, 256, 0, stream>>>(ffnb[0], out_x[0], ffnb[1], out_x[1], dw, acc);
  k_finalize<<<1, 32, 0, stream>>>(acc, out_loss);

  (void)in_sizes; (void)n_in; (void)out_size; (void)ws_size;
}